// ELangevinSampler_36507222016500
// MI455X (gfx1250) — compile-verified
//
#include <hip/hip_runtime.h>
#include <cstdint>
#include <cstddef>

// ---------------------------------------------------------------------------
// Discrete Langevin sampler, fused 5-step kernel for gfx1250 (MI455X).
//   B=16384 rows, D=256, NSTEPS=5.  Each WG owns 64 rows (4 waves x 16 rows),
//   keeps x (f16), x_a (f32), x_delta (f16), x_delta_a (f32) resident in LDS
//   for all 5 steps.  W is pre-swizzled into d_ws and streamed through a
//   double-buffered 16KB LDS block with GLOBAL_LOAD_ASYNC_TO_LDS_B128
//   (ASYNCcnt pipelined); GEMMs use V_WMMA_F32_16X16X4_F32 (exact fp32 math).
//   rr/noise stream with nontemporal loads.
// ---------------------------------------------------------------------------

#define Bc 16384
#define Dc 256
#define NSTEP 5

typedef float    v2f __attribute__((ext_vector_type(2)));
typedef float    v4f __attribute__((ext_vector_type(4)));
typedef float    v8f __attribute__((ext_vector_type(8)));
typedef _Float16 v2h __attribute__((ext_vector_type(2)));

struct alignas(16) H8 { _Float16 h[8]; };

// ---- LDS layout (dynamic) --------------------------------------------------
constexpr int WBUF_BYTES = 16384;              // one W k-block: 16 rows x 256 cols f32
constexpr int OFF_WBUF0  = 0;
constexpr int OFF_WBUF1  = WBUF_BYTES;
constexpr int OFF_B      = 2 * WBUF_BYTES;     // 256 f32 = 1KB
constexpr int OFF_WAVE   = OFF_B + 1024;
// per-wave region:
constexpr int PW_XS    = 0;        // 16x256 f16 = 8192
constexpr int PW_XDS   = 8192;     // 16x256 f16 = 8192
constexpr int PW_XAS   = 16384;    // 16x256 f32 = 16384
constexpr int PW_XDAS  = 32768;    // 16x256 f32 = 16384
constexpr int PW_AFLAG = 49152;    // 16 f32 = 64
constexpr int PW_RSUM  = 49216;    // 16 rows x 4 f32 = 256
constexpr int PW_SIZE  = 49472;
constexpr int SMEM_BYTES = OFF_WAVE + 4 * PW_SIZE;   // 231,680 bytes < 320KB

// ---------------------------------------------------------------------------
// Pre-pass: swizzle W so that the 4 K-values of each WMMA B-fragment are
// contiguous:  wsw[kb*4096 + j*1024 + n*4 + kk] = W[(kb*16 + j*4 + kk)*256 + n]
// ---------------------------------------------------------------------------
__global__ void swizzle_w_kernel(const float* __restrict__ W, float* __restrict__ wsw) {
  int idx = blockIdx.x * 256 + threadIdx.x;        // 0 .. 65535
  int k = idx >> 8, n = idx & 255;
  int kb = k >> 4, j = (k >> 2) & 3, kk = k & 3;
  wsw[kb * 4096 + j * 1024 + n * 4 + kk] = W[idx];
}

// ---------------------------------------------------------------------------
// Async copy of one 16KB W block into LDS: 128 threads x 8 x b128.
// IOFFSET is added to BOTH the global address and the LDS address, so the
// contiguous copy uses voff/ldsv = base + tid*16 and imm = i*2048.
// ---------------------------------------------------------------------------
__device__ __forceinline__ void issue_wblock_async(uint32_t ldsv, uint32_t voff,
                                                   uint64_t gbase) {
#pragma unroll
  for (int i = 0; i < 8; ++i)
    asm volatile("global_load_async_to_lds_b128 %0, %1, %2 offset:%c3"
                 :: "v"(ldsv), "v"(voff), "s"(gbase), "i"(i * 2048)
                 : "memory");
}

// ---------------------------------------------------------------------------
// One GEMM:  acc[16 tiles of 16x16] += xsrc(16x256, f16 in LDS) @ W(256x256).
// W streamed from global (swizzled) through double-buffered LDS blocks via
// async-to-LDS; all 4 waves of the WG participate (barriers per block).
// ---------------------------------------------------------------------------
__device__ __forceinline__ void gemm16(char* lds, const char* xsrc,
                                       const float* __restrict__ wsw, v8f acc[16]) {
  const int tid  = threadIdx.x;
  const int lane = tid & 31;
  const int lo   = lane & 15;
  const int hi   = lane >> 4;
  const uint64_t gbase = (uint64_t)(uintptr_t)wsw;
  // LDS aperture: low 32 bits of a generic LDS pointer == LDS byte offset.
  const uint32_t lds0 = (uint32_t)(uintptr_t)(lds + OFF_WBUF0) + tid * 16;
  const uint32_t lds1 = (uint32_t)(uintptr_t)(lds + OFF_WBUF1) + tid * 16;

  // kick off block 0 -> buf0   (ASYNCcnt += 8)
  issue_wblock_async(lds0, (uint32_t)(tid * 16), gbase);

  for (int kb = 0; kb < 16; ++kb) {
    __syncthreads();                       // all waves done computing block kb-1
    if (kb < 15) {
      // prefetch block kb+1 into the other buffer (ASYNCcnt: 8 -> 16)
      issue_wblock_async(((kb + 1) & 1) ? lds1 : lds0,
                         (uint32_t)(tid * 16 + (kb + 1) * WBUF_BYTES), gbase);
      // async loads retire in order: <=8 outstanding => block kb is in LDS
      asm volatile("s_wait_asynccnt 8" ::: "memory");
    } else {
      asm volatile("s_wait_asynccnt 0" ::: "memory");
    }
    __syncthreads();                       // block kb visible to all waves

    const char* wb = lds + ((kb & 1) ? OFF_WBUF1 : OFF_WBUF0);
#pragma unroll
    for (int j = 0; j < 4; ++j) {
      // A fragment 16x4 f32: lane -> (M = lo, K = kb*16 + j*4 + 2*hi + {0,1})
      v2h ah = *(const v2h*)(xsrc + lo * 512 + (kb * 16 + j * 4 + 2 * hi) * 2);
      v2f A;
      A.x = (float)ah.x;
      A.y = (float)ah.y;
      const char* wbase = wb + j * 4096 + lo * 16 + hi * 8;
#pragma unroll
      for (int nt = 0; nt < 16; ++nt) {
        // B fragment 4x16 f32: lane -> (K = .. + 2*hi + {0,1}, N = nt*16 + lo)
        v2f Bv = *(const v2f*)(wbase + nt * 256);
        acc[nt] = __builtin_amdgcn_wmma_f32_16x16x4_f32(
            false, A, false, Bv, (short)0, acc[nt], false, false);
      }
    }
  }
}

__device__ __forceinline__ float rowsum16(float v) {
  v += __shfl_xor(v, 1);
  v += __shfl_xor(v, 2);
  v += __shfl_xor(v, 4);
  v += __shfl_xor(v, 8);
  return v;   // sum over the 16 lanes of this half-wave
}

// ---------------------------------------------------------------------------
__global__ __launch_bounds__(128, 1) void sampler_kernel(
    const float* __restrict__ x_in, const float* __restrict__ xa_in,
    const float* __restrict__ wsw,  const float* __restrict__ bvec,
    const float* __restrict__ rr,   const float* __restrict__ nz,
    const float* __restrict__ au,
    float* __restrict__ x_out, float* __restrict__ xa_out) {
  extern __shared__ char lds[];
  const int tid  = threadIdx.x;
  const int wave = tid >> 5, lane = tid & 31, lo = lane & 15, hi = lane >> 4;
  const int m0g  = blockIdx.x * 64 + wave * 16;   // first global row of this wave

  char*      wv    = lds + OFF_WAVE + wave * PW_SIZE;
  _Float16*  xsH   = (_Float16*)(wv + PW_XS);
  _Float16*  xdsH  = (_Float16*)(wv + PW_XDS);
  float*     xasF  = (float*)(wv + PW_XAS);
  float*     xdasF = (float*)(wv + PW_XDAS);
  float*     afl   = (float*)(wv + PW_AFLAG);
  float*     rsum  = (float*)(wv + PW_RSUM);
  float*     bl    = (float*)(lds + OFF_B);

  // ---- init: bias + state staging -----------------------------------------
  bl[tid]       = bvec[tid];
  bl[tid + 128] = bvec[tid + 128];
#pragma unroll
  for (int r = 0; r < 16; ++r) {
    const v4f* sx = (const v4f*)(x_in + (size_t)(m0g + r) * Dc) + lane * 2;
    v4f a0 = sx[0], a1 = sx[1];
    H8 hh;
    hh.h[0] = (_Float16)a0.x; hh.h[1] = (_Float16)a0.y;
    hh.h[2] = (_Float16)a0.z; hh.h[3] = (_Float16)a0.w;
    hh.h[4] = (_Float16)a1.x; hh.h[5] = (_Float16)a1.y;
    hh.h[6] = (_Float16)a1.z; hh.h[7] = (_Float16)a1.w;
    *(H8*)((char*)xsH + r * 512 + lane * 16) = hh;
    const v4f* sa = (const v4f*)(xa_in + (size_t)(m0g + r) * Dc) + lane * 2;
    v4f* da = (v4f*)((char*)xasF + r * 1024) + lane * 2;
    da[0] = sa[0]; da[1] = sa[1];
  }
  __syncthreads();

  for (int s = 0; s < NSTEP; ++s) {
    // ============ GEMM1: t1 = x @ W ============
    v8f acc[16];
#pragma unroll
    for (int nt = 0; nt < 16; ++nt) acc[nt] = (v8f){0, 0, 0, 0, 0, 0, 0, 0};
    gemm16(lds, (const char*)xsH, wsw, acc);

    // ============ elementwise phase 1 ============
    float a_lpf[8], a_e1[8], a_s1[8], a_nq[8];
#pragma unroll
    for (int r = 0; r < 8; ++r) { a_lpf[r] = 0.f; a_e1[r] = 0.f; a_s1[r] = 0.f; a_nq[r] = 0.f; }
#pragma unroll
    for (int nt = 0; nt < 16; ++nt) {
      v8f cc = acc[nt];
      const int   col = nt * 16 + lo;
      const float bv  = bl[col];
#pragma unroll
      for (int r = 0; r < 8; ++r) {
        const int    rowl = r + hi * 8;
        const size_t gidx = ((size_t)s * Bc + (m0g + rowl)) * Dc + col;
        float xv  = (float)xsH[rowl * Dc + col];
        float xav = xasF[rowl * Dc + col];
        float rrv = __builtin_nontemporal_load(&rr[gidx]);
        float nv  = __builtin_nontemporal_load(&nz[gidx]);
        float t   = cc[r] + bv;                       // (xW + b)
        float ga  = (xv - xav) * 1.0e-4f;             // /ETA
        float sgn = 1.0f - 2.0f * xv;                 // -(2x-1)
        float gm  = (t - ga) * sgn * 0.5f;            // /TEMP
        float fp  = 1.0f / (1.0f + __expf(2.5f - gm));  // sigmoid(gm - 1/(2*ALPHA))
        bool  ind = rrv < fp;
        float xd  = ind ? (1.0f - xv) : xv;
        float xda = xav + 0.25f * ga + 0.70710678118654752f * nv;
        a_lpf[r] += __logf((ind ? fp : 1.0f - fp) + 1.0e-10f);
        a_e1[r]  += (t + bv) * xv;                    // 2*energy(x) pieces
        a_s1[r]  += (xv - xav) * (xv - xav);
        a_nq[r]  += nv * nv;                          // add_fwd = -0.5*sum(nv^2)
        xdsH[rowl * Dc + col]  = (_Float16)xd;
        xdasF[rowl * Dc + col] = xda;
      }
    }
    // reduce phase-1 sums and park them in LDS (frees VGPRs for GEMM2)
#pragma unroll
    for (int r = 0; r < 8; ++r) {
      float v0 = rowsum16(a_lpf[r]);
      float v1 = rowsum16(a_e1[r]);
      float v2 = rowsum16(a_s1[r]);
      float v3 = rowsum16(a_nq[r]);
      if (lo == 0) {
        const int rowl = r + hi * 8;
        rsum[rowl * 4 + 0] = v0; rsum[rowl * 4 + 1] = v1;
        rsum[rowl * 4 + 2] = v2; rsum[rowl * 4 + 3] = v3;
      }
    }

    // ============ GEMM2: t2 = x_delta @ W ============
#pragma unroll
    for (int nt = 0; nt < 16; ++nt) acc[nt] = (v8f){0, 0, 0, 0, 0, 0, 0, 0};
    gemm16(lds, (const char*)xdsH, wsw, acc);

    // ============ elementwise phase 2 ============
    float a_lpr[8], a_e2[8], a_s2[8], a_ar[8];
#pragma unroll
    for (int r = 0; r < 8; ++r) { a_lpr[r] = 0.f; a_e2[r] = 0.f; a_s2[r] = 0.f; a_ar[r] = 0.f; }
#pragma unroll
    for (int nt = 0; nt < 16; ++nt) {
      v8f cc = acc[nt];
      const int   col = nt * 16 + lo;
      const float bv  = bl[col];
#pragma unroll
      for (int r = 0; r < 8; ++r) {
        const int rowl = r + hi * 8;
        float xv   = (float)xsH[rowl * Dc + col];
        float xdv  = (float)xdsH[rowl * Dc + col];
        float xav  = xasF[rowl * Dc + col];
        float xdav = xdasF[rowl * Dc + col];
        bool  ind  = (xv != xdv);                     // flipped?
        float t    = cc[r] + bv;
        float ga2  = (xdv - xdav) * 1.0e-4f;
        float sgn  = 1.0f - 2.0f * xdv;
        float gm2  = (t - ga2) * sgn * 0.5f;
        float fp2  = 1.0f / (1.0f + __expf(2.5f - gm2));
        a_lpr[r] += __logf((ind ? fp2 : 1.0f - fp2) + 1.0e-10f);
        a_e2[r]  += (t + bv) * xdv;
        a_s2[r]  += (xdv - xdav) * (xdv - xdav);
        float dr = xav - (xdav + 0.25f * ga2);
        a_ar[r]  += dr * dr;
      }
    }

    // ============ MH accept ============
#pragma unroll
    for (int r = 0; r < 8; ++r) {
      float lpr = rowsum16(a_lpr[r]);
      float e2  = rowsum16(a_e2[r]);
      float s2  = rowsum16(a_s2[r]);
      float ar  = rowsum16(a_ar[r]);
      const int rowl = r + hi * 8;
      float lpf = rsum[rowl * 4 + 0];
      float e1  = rsum[rowl * 4 + 1];
      float s1  = rsum[rowl * 4 + 2];
      float nq  = rsum[rowl * 4 + 3];
      float m1  = 0.5f * e1 - 5.0e-5f * s1;           // energy - ssq/(2*ETA)
      float m2  = 0.5f * e2 - 5.0e-5f * s2;
      float la  = (m2 - m1) - ar + 0.5f * nq + lpr - lpf;
      float u   = au[(size_t)s * Bc + m0g + rowl];
      float af  = (la > __logf(u)) ? 1.0f : 0.0f;
      if (lo == 0) afl[rowl] = af;
    }
    // state update (per-wave staging; wave is lock-step, no WG barrier needed)
#pragma unroll
    for (int r = 0; r < 16; ++r) {
      float af = afl[r];
      if (af > 0.5f) {
        *(uint4*)((char*)xsH + r * 512 + lane * 16) =
            *(const uint4*)((char*)xdsH + r * 512 + lane * 16);
        v4f*       d  = (v4f*)((char*)xasF + r * 1024) + lane * 2;
        const v4f* sZ = (const v4f*)((char*)xdasF + r * 1024) + lane * 2;
        d[0] = sZ[0]; d[1] = sZ[1];
      }
    }
  }

  // ---- write results -------------------------------------------------------
#pragma unroll
  for (int r = 0; r < 16; ++r) {
    const H8 hh = *(const H8*)((char*)xsH + r * 512 + lane * 16);
    v4f o0, o1;
    o0.x = (float)hh.h[0]; o0.y = (float)hh.h[1]; o0.z = (float)hh.h[2]; o0.w = (float)hh.h[3];
    o1.x = (float)hh.h[4]; o1.y = (float)hh.h[5]; o1.z = (float)hh.h[6]; o1.w = (float)hh.h[7];
    float* dx = x_out + (size_t)(m0g + r) * Dc + lane * 8;
    __builtin_nontemporal_store(o0, (v4f*)dx);
    __builtin_nontemporal_store(o1, (v4f*)(dx + 4));
    const v4f* sa  = (const v4f*)((char*)xasF + r * 1024) + lane * 2;
    float* dxa = xa_out + (size_t)(m0g + r) * Dc + lane * 8;
    __builtin_nontemporal_store(sa[0], (v4f*)dxa);
    __builtin_nontemporal_store(sa[1], (v4f*)(dxa + 4));
  }
}

// ---------------------------------------------------------------------------
extern "C" void kernel_launch(void* const* d_in, const int* in_sizes, int n_in,
                              void* d_out, int out_size, void* d_ws, size_t ws_size,
                              hipStream_t stream) {
  (void)in_sizes; (void)n_in; (void)out_size; (void)ws_size;
  const float* x    = (const float*)d_in[0];
  const float* xa   = (const float*)d_in[1];
  const float* W    = (const float*)d_in[2];
  const float* bv   = (const float*)d_in[3];
  const float* rr   = (const float*)d_in[4];
  const float* nz   = (const float*)d_in[5];
  const float* au   = (const float*)d_in[6];
  float*       xout = (float*)d_out;
  float*       wsw  = (float*)d_ws;          // 256KB swizzled W

  (void)hipFuncSetAttribute((const void*)sampler_kernel,
                            hipFuncAttributeMaxDynamicSharedMemorySize, SMEM_BYTES);

  swizzle_w_kernel<<<256, 256, 0, stream>>>(W, wsw);
  sampler_kernel<<<Bc / 64, 128, SMEM_BYTES, stream>>>(
      x, xa, wsw, bv, rr, nz, au, xout, xout + (size_t)Bc * Dc);
}